// PAN_24309514896050
// MI455X (gfx1250) — compile-verified
//
#include <hip/hip_runtime.h>

// ---------------------------------------------------------------------------
// PAN (PANConv + PANPooling) pipeline for MI455X (gfx1250, wave32, WMMA+TDM).
// Sizes fixed by the reference:
//   G=128 graphs, N_PER=512, F=NHID=128, E=524288, L = 4/2/2, k = 256/128/64.
// All heavy scratch lives in d_ws (~150 MB) -> resident in the 192 MB L2.
// GEMMs: V_WMMA_F32_16X16X4_F32 (exact fp32 matrix pipe), A staged by the
// Tensor Data Mover into padded LDS, W pair-packed in LDS for direct
// ds_load_b64 fragments.
// ---------------------------------------------------------------------------

#define G_GRAPHS 128
#define NPER0    512
#define FEAT     128
#define E_TOT    524288
#define N0       (G_GRAPHS * NPER0)   // 65536

#define A_STRIDE 132                   // padded LDS row stride for A (floats)

typedef float v2f __attribute__((ext_vector_type(2)));
typedef float v8f __attribute__((ext_vector_type(8)));
typedef unsigned int u32x4 __attribute__((ext_vector_type(4)));
typedef int i32x8 __attribute__((ext_vector_type(8)));
typedef int i32x4 __attribute__((ext_vector_type(4)));

#ifndef __has_builtin
#define __has_builtin(x) 0
#endif
#if __has_builtin(__builtin_amdgcn_tensor_load_to_lds) && \
    __has_builtin(__builtin_amdgcn_s_wait_tensorcnt)
#define HAVE_TDM 1
#else
#define HAVE_TDM 0
#endif

__device__ __forceinline__ void atomAddF(float* p, float v) {
  __hip_atomic_fetch_add(p, v, __ATOMIC_RELAXED, __HIP_MEMORY_SCOPE_AGENT);
}

// ---------------- fp32 WMMA GEMM: C[n,128] = A[n,128] @ W[128,128] + bias ---
// Block = 256 threads = 8 waves; block does 128 rows, each wave a 16-row strip
// across all 8 column tiles.
//  - A block (128x128 f32) DMA'd by the TDM into LDS with 4-DWORD row padding
//    (pad_interval=128 DW, pad_amount=4 DW) -> conflict-free a-fragments.
//  - W pair-packed in LDS: element W[k][c] at  (k/2)*256 + ((2c) ^ ((k/2&1)<<5)) + (k&1)
//    so each B fragment {W[k][c], W[k+1][c]} is one aligned ds_load_b64, and the
//    XOR-32 swizzle puts the two lane halves in disjoint bank halves.
__global__ __launch_bounds__(256) void gemm128_wmma(
    const float* __restrict__ A, const float* __restrict__ W,
    const float* __restrict__ bias, float* __restrict__ C) {
  __shared__ float sA[128 * A_STRIDE];   // 66 KB, padded rows
  __shared__ float sW[128 * 128];        // 64 KB, pair-packed + swizzled

  const float* Ab = A + (size_t)blockIdx.x * 128 * FEAT;

  // pair-pack W (coalesced global reads; once per block)
  for (int i = threadIdx.x; i < 128 * 128; i += 256) {
    int k = i >> 7, c = i & 127;
    int p = k >> 1;
    sW[p * 256 + ((c * 2) ^ ((p & 1) << 5)) + (k & 1)] = W[i];
  }

#if HAVE_TDM
  if (threadIdx.x < 32) {  // one wave issues the tensor DMA for the A block
    unsigned long long ga = (unsigned long long)(const void*)Ab;
    unsigned lds = (unsigned)(unsigned long long)(void*)sA;
    u32x4 g0;
    g0[0] = 1u;                                   // count=1, user descriptor
    g0[1] = lds;                                  // lds_addr (bytes)
    g0[2] = (unsigned)ga;                         // global_addr[31:0]
    g0[3] = ((unsigned)(ga >> 32) & 0x01FFFFFFu)  // global_addr[56:32]
            | (2u << 30);                         // type=2 ("image")
    i32x8 g1;
    g1[0] = (int)((2u << 16)      // data_size = 4 bytes
                  | (1u << 20)    // pad_enable
                  | (6u << 22)    // pad_interval: 128 DWORDs
                  | (3u << 25));  // pad_amount:   4 DWORDs
    g1[1] = (int)(128u << 16);    // tensor_dim0[15:0]=128  (bits 79:48)
    g1[2] = (int)(128u << 16);    // tensor_dim1[15:0]=128  (bits 111:80)
    g1[3] = (int)(128u << 16);    // tile_dim0=128          (bits 127:112)
    g1[4] = 128;                  // tile_dim1=128          (bits 143:128)
    g1[5] = 128;                  // tensor_dim0_stride=128 (bits 207:160)
    g1[6] = 0;
    g1[7] = 0;                    // tensor_dim1_stride=0
    i32x4 z4;
    z4[0] = z4[1] = z4[2] = z4[3] = 0;
    i32x8 z8;
    for (int q = 0; q < 8; ++q) z8[q] = 0;
    // 6-arg form (amdgpu-toolchain clang-23): (g0, g1, g2, g3, g4, cpol)
    __builtin_amdgcn_tensor_load_to_lds(g0, g1, z4, z4, z8, 0);
    __builtin_amdgcn_s_wait_tensorcnt(0);
  }
#else
  for (int i = threadIdx.x; i < 128 * 128; i += 256) {
    int r = i >> 7, c = i & 127;
    sA[r * A_STRIDE + c] = Ab[i];
  }
#endif
  __syncthreads();

  const int wv    = threadIdx.x >> 5;
  const int lane  = threadIdx.x & 31;
  const int m0    = blockIdx.x * 128 + wv * 16;
  const int khalf = (lane >> 4) << 1;                 // 0 (lanes 0-15), 2 (16-31)
  const float* aptr = &sA[(wv * 16 + (lane & 15)) * A_STRIDE + khalf];
  const int l2 = (lane & 15) * 2;

  v8f acc[8];
#pragma unroll
  for (int t = 0; t < 8; ++t)
#pragma unroll
    for (int v = 0; v < 8; ++v) acc[t][v] = 0.f;

  for (int kk = 0; kk < 128; kk += 4) {
    // A fragment 16x4 (ISA layout: V0 = K0|K2, V1 = K1|K3 across lane halves)
    v2f a = *(const v2f*)(aptr + kk);
    const int p  = (kk + khalf) >> 1;
    const int sw = (p & 1) << 5;
    const float* wpt = &sW[p * 256 + l2];
#pragma unroll
    for (int t = 0; t < 8; ++t) {
      v2f b = *(const v2f*)(wpt + ((t * 32) ^ sw));  // {W[k][c], W[k+1][c]}
      acc[t] = __builtin_amdgcn_wmma_f32_16x16x4_f32(
          false, a, false, b, (short)0, acc[t], false, false);
    }
  }

  // C layout: VGPR v, lanes 0-15 -> M=v, N=lane; lanes 16-31 -> M=v+8, N=lane-16
  const int nb   = lane & 15;
  const int moff = (lane >> 4) * 8;
#pragma unroll
  for (int t = 0; t < 8; ++t) {
    const int col = t * 16 + nb;
    const float bv = bias[col];
#pragma unroll
    for (int v = 0; v < 8; ++v)
      C[(size_t)(m0 + moff + v) * FEAT + col] = acc[t][v] + bv;
  }
}

// ---------------- element-wise helpers -------------------------------------
__global__ void fill_f32(float* p, float v, int n) {
  int i = blockIdx.x * blockDim.x + threadIdx.x;
  if (i < n) p[i] = v;
}
__global__ void fill_i32(int* p, int v, int n) {
  int i = blockIdx.x * blockDim.x + threadIdx.x;
  if (i < n) p[i] = v;
}
__global__ void rsqrt_kernel(const float* d, float* o, int n) {
  int i = blockIdx.x * blockDim.x + threadIdx.x;
  if (i < n) o[i] = rsqrtf(fmaxf(d[i], 1e-12f));
}
// out1 (and optionally out2) = x * s[row]
__global__ void rowscale(const float* __restrict__ x, const float* __restrict__ s,
                         float* __restrict__ o1, float* o2, int n) {
  int i = blockIdx.x * blockDim.x + threadIdx.x;
  if (i < n) {
    float v = x[i] * s[i >> 7];
    o1[i] = v;
    if (o2) o2[i] = v;
  }
}
// acc += z; y = z; z = 0  (one pass of the pan_prop recurrence bookkeeping)
__global__ void prop_update(float* __restrict__ acc, float* __restrict__ y,
                            float* __restrict__ z, int n) {
  int i = blockIdx.x * blockDim.x + threadIdx.x;
  if (i < n) {
    float v = z[i];
    acc[i] += v;
    y[i] = v;
    z[i] = 0.f;
  }
}

// ---------------- edge kernels ---------------------------------------------
// scalar: z[dst] += y[src] * w
__global__ void edge_scatter_scalar(const float* __restrict__ y, const int* __restrict__ src,
                                    const int* __restrict__ dst, const float* __restrict__ ew,
                                    float* __restrict__ z, int E) {
  int e = blockIdx.x * blockDim.x + threadIdx.x;
  if (e >= E) return;
  float w = ew[e];
  if (w != 0.f) atomAddF(&z[dst[e]], y[src[e]] * w);
}
// deg[dst] += w
__global__ void edge_deg(const int* __restrict__ dst, const float* __restrict__ ew,
                         float* __restrict__ deg, int E) {
  int e = blockIdx.x * blockDim.x + threadIdx.x;
  if (e >= E) return;
  float w = ew[e];
  if (w != 0.f) atomAddF(&deg[dst[e]], w);
}
// features: one wave32 per edge, 4 floats/lane -> edge metadata is wave-uniform
__global__ void edge_scatter_feat(const float* __restrict__ y, const int* __restrict__ src,
                                  const int* __restrict__ dst, const float* __restrict__ ew,
                                  float* __restrict__ z, int E) {
  int t = blockIdx.x * blockDim.x + threadIdx.x;
  int e = t >> 5;
  if (e >= E) return;
  float w = ew[e];
  if (w == 0.f) return;
  int f = (t & 31) << 2;
  const float4 v = *(const float4*)(y + (size_t)src[e] * FEAT + f);
  float* zp = z + (size_t)dst[e] * FEAT + f;
  atomAddF(zp + 0, v.x * w);
  atomAddF(zp + 1, v.y * w);
  atomAddF(zp + 2, v.z * w);
  atomAddF(zp + 3, v.w * w);
}

// ---------------- pooling kernels ------------------------------------------
__global__ void pnorm_kernel(const float* p, float* out) {
  __shared__ float s[128];
  int t = threadIdx.x;
  float v = p[t];
  s[t] = v * v;
  __syncthreads();
  for (int o = 64; o > 0; o >>= 1) {
    if (t < o) s[t] += s[t + o];
    __syncthreads();
  }
  if (t == 0) out[0] = sqrtf(s[0]);
}
__global__ void score_kernel(const float* __restrict__ h, const float* __restrict__ p,
                             const float* __restrict__ deg, const float* __restrict__ beta,
                             const float* __restrict__ pn, float* __restrict__ sc, int n) {
  int i = blockIdx.x * blockDim.x + threadIdx.x;
  if (i >= n) return;
  const float* hr = h + (size_t)i * FEAT;
  float acc = 0.f;
  for (int k = 0; k < FEAT; k += 4) {
    float4 hv = *(const float4*)(hr + k);
    float4 pv = *(const float4*)(p + k);
    acc += hv.x * pv.x + hv.y * pv.y + hv.z * pv.z + hv.w * pv.w;
  }
  sc[i] = beta[0] * acc / pn[0] + beta[1] * deg[i];
}
// per-graph bitonic top-k (n_per is a power of two <= 512); one block per graph
__global__ void topk_kernel(const float* __restrict__ score, int n_per, int k,
                            int* __restrict__ perm) {
  __shared__ float sc[512];
  __shared__ int   id[512];
  int g = blockIdx.x;
  int base = g * n_per;
  for (int i = threadIdx.x; i < n_per; i += blockDim.x) {
    sc[i] = score[base + i];
    id[i] = i;
  }
  __syncthreads();
  for (int size = 2; size <= n_per; size <<= 1) {
    for (int stride = size >> 1; stride > 0; stride >>= 1) {
      for (int i = threadIdx.x; i < (n_per >> 1); i += blockDim.x) {
        int pos = 2 * i - (i & (stride - 1));
        int a = pos, b = pos + stride;
        bool desc = ((pos & size) == 0);
        float sa = sc[a], sb = sc[b];
        int ia = id[a], ib = id[b];
        bool aFirst = (sa > sb) || (sa == sb && ia < ib);  // descending, idx tiebreak
        if (desc ? !aFirst : aFirst) {
          sc[a] = sb; sc[b] = sa;
          id[a] = ib; id[b] = ia;
        }
      }
      __syncthreads();
    }
  }
  for (int i = threadIdx.x; i < k; i += blockDim.x) perm[g * k + i] = base + id[i];
}
// x_new[i] = h[perm[i]] * tanh(score[perm[i]]) ; one wave per row
__global__ void gather_scale(const float* __restrict__ h, const int* __restrict__ perm,
                             const float* __restrict__ score, float* __restrict__ out, int M) {
  int t = blockIdx.x * blockDim.x + threadIdx.x;
  int i = t >> 5;
  if (i >= M) return;
  int r = perm[i];
  float s = tanhf(score[r]);
  int f = (t & 31) << 2;
  const float4 v = *(const float4*)(h + (size_t)r * FEAT + f);
  float4 o;
  o.x = v.x * s; o.y = v.y * s; o.z = v.z * s; o.w = v.w * s;
  *(float4*)(out + (size_t)i * FEAT + f) = o;
}
__global__ void inv_mask_kernel(const int* __restrict__ perm, int M,
                                int* __restrict__ inv, float* __restrict__ mask) {
  int i = blockIdx.x * blockDim.x + threadIdx.x;
  if (i < M) {
    int r = perm[i];
    inv[r] = i;
    mask[r] = 1.f;
  }
}
__global__ void remap_edges(const int* __restrict__ src, const int* __restrict__ dst,
                            const float* __restrict__ ew, const int* __restrict__ inv,
                            const float* __restrict__ mask, int* __restrict__ src2,
                            int* __restrict__ dst2, float* __restrict__ ew2, int E) {
  int e = blockIdx.x * blockDim.x + threadIdx.x;
  if (e >= E) return;
  int s = src[e], d = dst[e];
  src2[e] = inv[s];
  dst2[e] = inv[d];
  ew2[e]  = ew[e] * mask[s] * mask[d];
}

// ---------------- readout --------------------------------------------------
__global__ void mean_kernel(const float* __restrict__ h, float* __restrict__ mean, int kper) {
  int g = blockIdx.x, f = threadIdx.x;  // 128 threads
  float s = 0.f;
  for (int j = 0; j < kper; ++j) s += h[(size_t)(g * kper + j) * FEAT + f];
  mean[g * FEAT + f] = s / (float)kper;
}
__global__ void head_kernel(const float* __restrict__ mean, const float* __restrict__ W1,
                            const float* __restrict__ b1, const float* __restrict__ W2,
                            const float* __restrict__ b2, float* __restrict__ out) {
  __shared__ float s[64];
  int g = blockIdx.x, j = threadIdx.x;  // 64 threads
  float a = b1[j];
  for (int k = 0; k < FEAT; ++k) a += mean[g * FEAT + k] * W1[k * 64 + j];
  a = a > 0.f ? a : 0.f;
  s[j] = a * W2[j];
  __syncthreads();
  for (int o = 32; o > 0; o >>= 1) {
    if (j < o) s[j] += s[j + o];
    __syncthreads();
  }
  if (j == 0) out[g] = s[0] + b2[0];
}

// ---------------------------------------------------------------------------
static inline int cdiv(int a, int b) { return (a + b - 1) / b; }

extern "C" void kernel_launch(void* const* d_in, const int* in_sizes, int n_in,
                              void* d_out, int out_size, void* d_ws, size_t ws_size,
                              hipStream_t stream) {
  (void)in_sizes; (void)n_in; (void)out_size; (void)ws_size;
  const float* x       = (const float*)d_in[0];
  const int*   ei      = (const int*)d_in[1];      // [2, E]: src then dst
  const float* lump_W  = (const float*)d_in[3];
  const float* lump_b  = (const float*)d_in[4];
  const float* conv1_W = (const float*)d_in[5];
  const float* conv1_b = (const float*)d_in[6];
  const float* conv2_W = (const float*)d_in[7];
  const float* conv2_b = (const float*)d_in[8];
  const float* conv3_W = (const float*)d_in[9];
  const float* conv3_b = (const float*)d_in[10];
  const float* p1 = (const float*)d_in[11]; const float* be1 = (const float*)d_in[12];
  const float* p2 = (const float*)d_in[13]; const float* be2 = (const float*)d_in[14];
  const float* p3 = (const float*)d_in[15]; const float* be3 = (const float*)d_in[16];
  const float* lin1_W = (const float*)d_in[17];
  const float* lin1_b = (const float*)d_in[18];
  const float* lin2_W = (const float*)d_in[19];
  const float* lin2_b = (const float*)d_in[20];
  float* out = (float*)d_out;

  const int* src0 = ei;
  const int* dst0 = ei + E_TOT;

  // ---- workspace carve (bump allocator, 256 B aligned) ----
  char* wp = (char*)d_ws;
  auto take = [&](size_t bytes) -> void* {
    char* r = wp;
    wp += (bytes + 255) & ~(size_t)255;
    return (void*)r;
  };
  const size_t FB = (size_t)N0 * FEAT * sizeof(float);
  float* b0 = (float*)take(FB);   // current node features (conv input / pooled x)
  float* b1 = (float*)take(FB);   // prop y / conv output
  float* b2 = (float*)take(FB);   // prop z / gemm input (hpre)
  float* b3 = (float*)take(FB);   // prop acc
  float* dvec  = (float*)take((size_t)N0 * 4);
  float* dinv  = (float*)take((size_t)N0 * 4);
  float* degv  = (float*)take((size_t)N0 * 4);
  float* score = (float*)take((size_t)N0 * 4);
  float* sy    = (float*)take((size_t)N0 * 4);
  float* sz    = (float*)take((size_t)N0 * 4);
  float* mask  = (float*)take((size_t)N0 * 4);
  int*   inv   = (int*)  take((size_t)N0 * 4);
  int*   perm  = (int*)  take((size_t)(N0 / 2) * 4);
  float* ew0   = (float*)take((size_t)E_TOT * 4);
  int*   srcA  = (int*)  take((size_t)E_TOT * 4);
  int*   dstA  = (int*)  take((size_t)E_TOT * 4);
  float* ewA   = (float*)take((size_t)E_TOT * 4);
  int*   srcB  = (int*)  take((size_t)E_TOT * 4);
  int*   dstB  = (int*)  take((size_t)E_TOT * 4);
  float* ewB   = (float*)take((size_t)E_TOT * 4);
  float* pnorm = (float*)take(256);
  float* meanb = (float*)take((size_t)G_GRAPHS * FEAT * 4);

  auto fillF = [&](float* p, float v, int n) {
    fill_f32<<<cdiv(n, 256), 256, 0, stream>>>(p, v, n);
  };
  auto fillI = [&](int* p, int v, int n) {
    fill_i32<<<cdiv(n, 256), 256, 0, stream>>>(p, v, n);
  };

  // PANConv: b0 (N x 128) -> b1, using edges (s,d,w), filter order L.
  auto conv = [&](int Nn, int L, const int* s, const int* d, const float* w,
                  const float* Wm, const float* bb) {
    // degree part: dvec = sum_{l<=L} A^l * 1
    fillF(sy, 1.f, Nn);
    fillF(dvec, 1.f, Nn);
    fillF(sz, 0.f, Nn);
    for (int l = 0; l < L; ++l) {
      edge_scatter_scalar<<<cdiv(E_TOT, 256), 256, 0, stream>>>(sy, s, d, w, sz, E_TOT);
      prop_update<<<cdiv(Nn, 256), 256, 0, stream>>>(dvec, sy, sz, Nn);
    }
    rsqrt_kernel<<<cdiv(Nn, 256), 256, 0, stream>>>(dvec, dinv, Nn);
    // feature part: acc = sum_{l<=L} A^l (dinv .* x)
    int nf = Nn * FEAT;
    rowscale<<<cdiv(nf, 256), 256, 0, stream>>>(b0, dinv, b1, b3, nf);  // y + acc
    fillF(b2, 0.f, nf);                                                 // z
    for (int l = 0; l < L; ++l) {
      edge_scatter_feat<<<cdiv(E_TOT * 32, 256), 256, 0, stream>>>(b1, s, d, w, b2, E_TOT);
      prop_update<<<cdiv(nf, 256), 256, 0, stream>>>(b3, b1, b2, nf);
    }
    rowscale<<<cdiv(nf, 256), 256, 0, stream>>>(b3, dinv, b2, nullptr, nf);  // hpre
    gemm128_wmma<<<Nn / 128, 256, 0, stream>>>(b2, Wm, bb, b1);
  };

  // PANPooling: h = b1 (N x 128) -> x_new = b0 (G*k x 128); optionally remap edges.
  auto pool = [&](int Nn, int nper, int k, const float* pv, const float* bev,
                  const int* s, const int* d, const float* w,
                  int* s2, int* d2, float* w2, bool remap) {
    fillF(degv, 0.f, Nn);
    edge_deg<<<cdiv(E_TOT, 256), 256, 0, stream>>>(d, w, degv, E_TOT);
    pnorm_kernel<<<1, 128, 0, stream>>>(pv, pnorm);
    score_kernel<<<cdiv(Nn, 256), 256, 0, stream>>>(b1, pv, degv, bev, pnorm, score, Nn);
    topk_kernel<<<G_GRAPHS, 256, 0, stream>>>(score, nper, k, perm);
    int M = G_GRAPHS * k;
    gather_scale<<<cdiv(M * 32, 256), 256, 0, stream>>>(b1, perm, score, b0, M);
    if (remap) {
      fillI(inv, 0, Nn);
      fillF(mask, 0.f, Nn);
      inv_mask_kernel<<<cdiv(M, 256), 256, 0, stream>>>(perm, M, inv, mask);
      remap_edges<<<cdiv(E_TOT, 256), 256, 0, stream>>>(s, d, w, inv, mask, s2, d2, w2, E_TOT);
    }
  };

  // ---------------- pipeline ----------------
  fillF(ew0, 1.f, E_TOT);
  gemm128_wmma<<<N0 / 128, 256, 0, stream>>>(x, lump_W, lump_b, b0);      // PANLump

  conv(N0, 4, src0, dst0, ew0, conv1_W, conv1_b);                         // conv1 -> b1
  pool(N0, NPER0, 256, p1, be1, src0, dst0, ew0, srcA, dstA, ewA, true);  // -> b0

  const int N2 = G_GRAPHS * 256;
  conv(N2, 2, srcA, dstA, ewA, conv2_W, conv2_b);                         // conv2 -> b1
  pool(N2, 256, 128, p2, be2, srcA, dstA, ewA, srcB, dstB, ewB, true);    // -> b0

  const int N3 = G_GRAPHS * 128;
  conv(N3, 2, srcB, dstB, ewB, conv3_W, conv3_b);                         // conv3 -> b1
  pool(N3, 128, 64, p3, be3, srcB, dstB, ewB, nullptr, nullptr, nullptr,
       false);                                                            // -> b0

  mean_kernel<<<G_GRAPHS, 128, 0, stream>>>(b0, meanb, 64);               // scatter_mean
  head_kernel<<<G_GRAPHS, 64, 0, stream>>>(meanb, lin1_W, lin1_b, lin2_W, lin2_b, out);
}